// SAGEEncoder_21431886807833
// MI455X (gfx1250) — compile-verified
//
#include <hip/hip_runtime.h>

#define NNODES 50000
#define NEDGES 800000
#define DIN    128      // feature dim entering both layers (D_IN == H1 == 128)
#define H1     128
#define H2     64

typedef __attribute__((ext_vector_type(2))) float v2f;
typedef __attribute__((ext_vector_type(8))) float v8f;

// ---------------------------------------------------------------- utilities
__global__ void zero_f32(float* __restrict__ p, int n) {
    int i = blockIdx.x * blockDim.x + threadIdx.x;
    if (i < n) p[i] = 0.0f;
}

__global__ void degree_k(const int* __restrict__ dst, float* __restrict__ deg, int E) {
    int e = blockIdx.x * blockDim.x + threadIdx.x;
    if (e < E) atomicAdd(&deg[dst[e]], 1.0f);
}

// One wave32 per edge: lane moves float4 of the 128-float row (coalesced
// global_load_b128 of x[src]) and atomically accumulates into agg[dst] (L2).
__global__ void scatter_k(const float* __restrict__ feat,
                          const int* __restrict__ src,
                          const int* __restrict__ dst,
                          float* __restrict__ agg, int E) {
    int warp = (blockIdx.x * blockDim.x + threadIdx.x) >> 5;
    int lane = threadIdx.x & 31;
    if (warp >= E) return;
    const int s = src[warp];
    const int d = dst[warp];
    const float4 v = *(const float4*)(feat + (size_t)s * DIN + lane * 4);
    float* ap = agg + (size_t)d * DIN + lane * 4;
    atomicAdd(ap + 0, v.x);
    atomicAdd(ap + 1, v.y);
    atomicAdd(ap + 2, v.z);
    atomicAdd(ap + 3, v.w);
}

// ------------------------------------------------------- fused SAGE GEMM
// out[m, n] = act( (agg[m,:]/max(deg,1)) @ Wl  +  bl  +  x[m,:] @ Wr )
// One wave computes one 16x16 tile with V_WMMA_F32_16X16X4_F32.
// Block = (DOUT/16) waves; each wave owns one 16-wide N-tile.
// A 16x4 layout: lane l (l16=l&15 -> M), half=l>>4; reg j holds K = k + 2*half + j.
// B 4x16 layout (symmetric): reg j holds K = k + 2*half + j, N = n_base + l16.
// C/D layout: reg r -> M = m_base + r + 8*half, N = n_base + l16.
template <int DOUT, bool RELU>
__global__ void sage_gemm_wmma(const float* __restrict__ agg,
                               const float* __restrict__ deg,
                               const float* __restrict__ xr,
                               const float* __restrict__ Wl,
                               const float* __restrict__ bl,
                               const float* __restrict__ Wr,
                               float* __restrict__ out) {
    const int lane   = threadIdx.x & 31;
    const int wave   = threadIdx.x >> 5;       // N-tile index
    const int m_base = blockIdx.x * 16;
    const int n_base = wave * 16;
    const int half   = lane >> 4;              // 0: lanes 0-15, 1: lanes 16-31
    const int l16    = lane & 15;

    const int   row    = m_base + l16;                 // A-row owned by this lane
    const float invdeg = 1.0f / fmaxf(deg[row], 1.0f); // mean-aggregation scale

    // init accumulator with bias (depends only on N -> same value in all 8 regs)
    const float b = bl[n_base + l16];
    v8f c;
#pragma unroll
    for (int i = 0; i < 8; ++i) c[i] = b;

    const float* arow = agg + (size_t)row * DIN;
    const float* xrow = xr  + (size_t)row * DIN;

    // chain 1: (agg * invdeg) @ Wl
#pragma unroll
    for (int k = 0; k < DIN; k += 4) {
        const int ka = k + 2 * half;
        v2f a;
        a.x = arow[ka + 0] * invdeg;
        a.y = arow[ka + 1] * invdeg;
        v2f bm;
        bm.x = Wl[(size_t)(ka + 0) * DOUT + n_base + l16];
        bm.y = Wl[(size_t)(ka + 1) * DOUT + n_base + l16];
        c = __builtin_amdgcn_wmma_f32_16x16x4_f32(false, a, false, bm,
                                                  (short)0, c, false, false);
    }

    // chain 2: x @ Wr
#pragma unroll
    for (int k = 0; k < DIN; k += 4) {
        const int ka = k + 2 * half;
        v2f a = *(const v2f*)&xrow[ka];           // 8B-aligned contiguous pair
        v2f bm;
        bm.x = Wr[(size_t)(ka + 0) * DOUT + n_base + l16];
        bm.y = Wr[(size_t)(ka + 1) * DOUT + n_base + l16];
        c = __builtin_amdgcn_wmma_f32_16x16x4_f32(false, a, false, bm,
                                                  (short)0, c, false, false);
    }

    // store D: reg r -> row m_base + r + 8*half, col n_base + l16
#pragma unroll
    for (int r = 0; r < 8; ++r) {
        float v = c[r];
        if (RELU) v = fmaxf(v, 0.0f);
        out[(size_t)(m_base + r + 8 * half) * DOUT + n_base + l16] = v;
    }
}

// ---------------------------------------------------------------- launcher
extern "C" void kernel_launch(void* const* d_in, const int* in_sizes, int n_in,
                              void* d_out, int out_size, void* d_ws, size_t ws_size,
                              hipStream_t stream) {
    const float* x   = (const float*)d_in[0];   // [N, 128]
    const float* Wl1 = (const float*)d_in[1];   // [128, 128]
    const float* bl1 = (const float*)d_in[2];   // [128]
    const float* Wr1 = (const float*)d_in[3];   // [128, 128]
    const float* Wl2 = (const float*)d_in[4];   // [128, 64]
    const float* bl2 = (const float*)d_in[5];   // [64]
    const float* Wr2 = (const float*)d_in[6];   // [128, 64]
    const int*   ei  = (const int*)d_in[7];     // [2, E]
    const int*   src = ei;
    const int*   dst = ei + NEDGES;
    float*       out = (float*)d_out;           // [N, 64]

    // workspace: deg[N] | agg[N*128] | h[N*128]
    float* deg = (float*)d_ws;
    float* agg = deg + NNODES;
    float* h   = agg + (size_t)NNODES * DIN;

    const int ZT = 256;
    const int aggN = NNODES * DIN;

    // degree (shared by both layers)
    zero_f32<<<(NNODES + ZT - 1) / ZT, ZT, 0, stream>>>(deg, NNODES);
    degree_k<<<(NEDGES + ZT - 1) / ZT, ZT, 0, stream>>>(dst, deg, NEDGES);

    // ---- layer 1 ----
    zero_f32<<<(aggN + ZT - 1) / ZT, ZT, 0, stream>>>(agg, aggN);
    scatter_k<<<NEDGES / 8, 256, 0, stream>>>(x, src, dst, agg, NEDGES);
    // 8 waves/block -> full 128-wide output row tile; 50000/16 = 3125 blocks
    sage_gemm_wmma<H1, true><<<NNODES / 16, 8 * 32, 0, stream>>>(
        agg, deg, x, Wl1, bl1, Wr1, h);

    // ---- layer 2 ----
    zero_f32<<<(aggN + ZT - 1) / ZT, ZT, 0, stream>>>(agg, aggN);
    scatter_k<<<NEDGES / 8, 256, 0, stream>>>(h, src, dst, agg, NEDGES);
    // 4 waves/block -> 64-wide output
    sage_gemm_wmma<H2, false><<<NNODES / 16, 4 * 32, 0, stream>>>(
        agg, deg, h, Wl2, bl2, Wr2, out);
}